// AttnDecoderRNN_82514911691263
// MI455X (gfx1250) — compile-verified
//
#include <hip/hip_runtime.h>
#include <math.h>

// Sizes from the reference
#define H_DIM 1024
#define V_DIM 50257
#define L_DIM 26

#define LS_BLOCKS 128

typedef __attribute__((ext_vector_type(2))) float v2f;
typedef __attribute__((ext_vector_type(8))) float v8f;

// ---------------------------------------------------------------------------
// Wave32 reductions
// ---------------------------------------------------------------------------
__device__ inline float wave_sum(float v) {
#pragma unroll
  for (int off = 16; off > 0; off >>= 1) v += __shfl_xor(v, off, 32);
  return v;
}

// ---------------------------------------------------------------------------
// WMMA GEMV core: 16 rows of W per wave, V_WMMA_F32_16X16X4_F32 per K-step.
// A layout (ISA 7.12.2, 16x4 f32): lanes 0-15 rows M=0..15 K={k,k+1};
// lanes 16-31 same rows K={k+2,k+3} -> lane l / l+16 read adjacent 8B of one
// row, so full 64B lines are consumed within a 4-step window.
// B = x broadcast into all 16 columns -> every column of D equals y.
// ---------------------------------------------------------------------------
__device__ inline v8f wmma_dot16(const float* __restrict__ wrow,
                                 const float* __restrict__ xs,
                                 int K, int half)
{
  v8f c = {0.f, 0.f, 0.f, 0.f, 0.f, 0.f, 0.f, 0.f};
  for (int k = 0; k < K; k += 4) {
    const int kk = k + half * 2;
    v2f a, b;
    a.x = wrow[kk];                     // merges to global_load_b64
    a.y = wrow[kk + 1];
    b.x = xs[kk];                       // merges to ds_load_b64
    b.y = xs[kk + 1];
    c = __builtin_amdgcn_wmma_f32_16x16x4_f32(
        /*neg_a=*/false, a, /*neg_b=*/false, b,
        /*c_mod=*/(short)0, c, /*reuse_a=*/false, /*reuse_b=*/false);
  }
  return c;
}

// D layout: VGPR j: lanes 0-15 hold M=m0+j, lanes 16-31 hold M=m0+8+j.
// All columns identical (B broadcast) -> lanes 0 and 16 write 8 results each.
__device__ inline void store_tile16(const v8f& c, int m0, int half, int lane,
                                    const float* __restrict__ bias,
                                    float* __restrict__ y, int M, int mode)
{
  if ((lane & 15) == 0) {
    const int base = m0 + half * 8;
#pragma unroll
    for (int j = 0; j < 8; ++j) {
      const int m = base + j;
      if (m < M) {
        float v = c[j] + bias[m];
        if (mode == 1) v = fmaxf(v, 0.f);
        y[m] = v;
      }
    }
  }
}

// ---------------------------------------------------------------------------
// Kernel A: embedding gather, attention logits + softmax, context vector.
// cat2[0..H) = embedded, cat2[H..2H) = attn_applied.  One block, 256 threads.
// ---------------------------------------------------------------------------
__global__ void __launch_bounds__(256)
attn_kernel(const int* __restrict__ token,
            const float* __restrict__ hidden,       // [H]
            const float* __restrict__ enc_out,      // [L,H]
            const float* __restrict__ emb,          // [V,H]
            const float* __restrict__ attn_W,       // [L,2H]
            const float* __restrict__ attn_b,       // [L]
            float* __restrict__ cat2)               // [2H] ws
{
  __shared__ float s_logit[32];
  __shared__ float s_w[32];

  const int tid  = threadIdx.x;
  const int lane = tid & 31;
  const int wave = tid >> 5;

  const int t = token[0];
  const float* erow = emb + (long)t * H_DIM;

  for (int i = tid; i < H_DIM; i += 256) cat2[i] = erow[i];

  for (int r = wave; r < L_DIM; r += 8) {
    const float* wr = attn_W + (long)r * (2 * H_DIM);
    float acc = 0.f;
    for (int k = lane; k < 2 * H_DIM; k += 32) {
      float c1 = (k < H_DIM) ? erow[k] : hidden[k - H_DIM];
      acc = fmaf(c1, wr[k], acc);
    }
    acc = wave_sum(acc);
    if (lane == 0) s_logit[r] = acc + attn_b[r];
  }
  __syncthreads();

  if (tid == 0) {                       // softmax over 26: trivial
    float m = -INFINITY;
    for (int l = 0; l < L_DIM; ++l) m = fmaxf(m, s_logit[l]);
    float s = 0.f;
    for (int l = 0; l < L_DIM; ++l) { float e = expf(s_logit[l] - m); s_w[l] = e; s += e; }
    const float inv = 1.f / s;
    for (int l = 0; l < L_DIM; ++l) s_w[l] *= inv;
  }
  __syncthreads();

  for (int h = tid; h < H_DIM; h += 256) {
    float s = 0.f;
#pragma unroll 2
    for (int l = 0; l < L_DIM; ++l) s = fmaf(s_w[l], enc_out[l * H_DIM + h], s);
    cat2[H_DIM + h] = s;
  }
}

// ---------------------------------------------------------------------------
// Generic WMMA GEMV: y = act(W @ x + bias).  128 threads = 4 waves = 64 rows
// per block (small blocks -> more blocks -> spread over more WGPs).
// mode: 0 = bias only, 1 = relu.
// ---------------------------------------------------------------------------
__global__ void __launch_bounds__(128)
gemv_wmma(const float* __restrict__ W, const float* __restrict__ x,
          const float* __restrict__ bias, float* __restrict__ y,
          int M, int K, int mode)
{
  __shared__ float xs[2048];            // K <= 2048 in this model
  {                                     // float4 staging -> global_load_b128
    const float4* x4  = (const float4*)x;
    float4*       xs4 = (float4*)xs;
    for (int i = threadIdx.x; i < (K >> 2); i += 128) xs4[i] = x4[i];
  }
  __syncthreads();

  const int lane = threadIdx.x & 31;
  const int wave = threadIdx.x >> 5;
  const int tile = blockIdx.x * 4 + wave;
  const int m0   = tile * 16;
  if (m0 >= M) return;                  // wave-uniform: EXEC stays full for WMMA

  const int half = lane >> 4;
  int row = m0 + (lane & 15);
  if (row >= M) row = M - 1;            // clamp loads on ragged last tile
  const float* wrow = W + (long)row * K;

  const v8f c = wmma_dot16(wrow, xs, K, half);
  store_tile16(c, m0, half, lane, bias, y, M, mode);
}

// ---------------------------------------------------------------------------
// Fused GRU-gate GEMVs: gi = w_ih@x + b_ih and gh = w_hh@h + b_hh in ONE
// launch (virtual M = 6144 rows, wave-uniform matrix select) -> 96 blocks
// concurrent instead of two serialized 48-block launches.
// ---------------------------------------------------------------------------
__global__ void __launch_bounds__(128)
gates_gemv(const float* __restrict__ w_ih, const float* __restrict__ x_in,
           const float* __restrict__ b_ih,
           const float* __restrict__ w_hh, const float* __restrict__ h_in,
           const float* __restrict__ b_hh,
           float* __restrict__ gi, float* __restrict__ gh)
{
  __shared__ float xs[2048];            // [0..1024) x_in, [1024..2048) h_in
  {
    const float4* a4 = (const float4*)x_in;
    const float4* b4 = (const float4*)h_in;
    float4*       s4 = (float4*)xs;
    for (int i = threadIdx.x; i < 256; i += 128) {
      s4[i]       = a4[i];
      s4[256 + i] = b4[i];
    }
  }
  __syncthreads();

  const int lane = threadIdx.x & 31;
  const int wave = threadIdx.x >> 5;
  const int tile = blockIdx.x * 4 + wave;   // 384 tiles total
  const bool second = tile >= 192;          // 3072/16 tiles per matrix

  const float* W    = second ? w_hh : w_ih;
  const float* bias = second ? b_hh : b_ih;
  float*       y    = second ? gh   : gi;
  const float* xsl  = xs + (second ? H_DIM : 0);
  const int    m0   = (second ? tile - 192 : tile) * 16;

  const int half = lane >> 4;
  const int row  = m0 + (lane & 15);        // 3072 % 16 == 0: no clamp needed
  const float* wrow = W + (long)row * H_DIM;

  const v8f c = wmma_dot16(wrow, xsl, H_DIM, half);
  store_tile16(c, m0, half, lane, bias, y, 3 * H_DIM, 0);
}

// ---------------------------------------------------------------------------
// GRU cell elementwise combine (PyTorch r,z,n layout).
// ---------------------------------------------------------------------------
__global__ void __launch_bounds__(256)
gru_kernel(const float* __restrict__ gi, const float* __restrict__ gh,
           const float* __restrict__ hidden,
           float* __restrict__ hnew_ws, float* __restrict__ hnew_out)
{
  const int h = blockIdx.x * blockDim.x + threadIdx.x;
  if (h >= H_DIM) return;
  const float r = 1.f / (1.f + expf(-(gi[h]         + gh[h])));
  const float z = 1.f / (1.f + expf(-(gi[H_DIM + h] + gh[H_DIM + h])));
  const float n = tanhf(gi[2 * H_DIM + h] + r * gh[2 * H_DIM + h]);
  const float o = (1.f - z) * n + z * hidden[h];
  hnew_ws[h]  = o;
  hnew_out[h] = o;
}

// ---------------------------------------------------------------------------
// Deterministic multi-block log-softmax (online m/s pairs, fixed combine
// order everywhere -> bitwise reproducible across replays).
// ---------------------------------------------------------------------------
__global__ void __launch_bounds__(256)
ls_partial(const float* __restrict__ logits, int n, float2* __restrict__ part)
{
  __shared__ float sm[8], ss[8];
  const int tid  = threadIdx.x;
  const int lane = tid & 31;
  const int wave = tid >> 5;

  float m = -INFINITY, s = 0.f;
  for (int i = blockIdx.x * 256 + tid; i < n; i += 256 * LS_BLOCKS) {
    const float xv = logits[i];
    if (xv > m) { s = s * expf(m - xv) + 1.f; m = xv; }
    else        { s += expf(xv - m); }
  }
  // wave combine (fixed xor pattern). Every lane saw >=1 element (n > 32768).
#pragma unroll
  for (int off = 16; off > 0; off >>= 1) {
    const float m2 = __shfl_xor(m, off, 32);
    const float s2 = __shfl_xor(s, off, 32);
    const float mm = fmaxf(m, m2);
    s = s * expf(m - mm) + s2 * expf(m2 - mm);
    m = mm;
  }
  if (lane == 0) { sm[wave] = m; ss[wave] = s; }
  __syncthreads();
  if (tid == 0) {                       // combine 8 wave partials, fixed order
    float mb = sm[0], sb = ss[0];
    for (int w = 1; w < 8; ++w) {
      const float mm = fmaxf(mb, sm[w]);
      sb = sb * expf(mb - mm) + ss[w] * expf(sm[w] - mm);
      mb = mm;
    }
    part[blockIdx.x] = make_float2(mb, sb);
  }
}

__global__ void ls_final(const float2* __restrict__ part, float* __restrict__ lse)
{
  if (threadIdx.x == 0 && blockIdx.x == 0) {
    float m = -INFINITY, s = 0.f;
    for (int i = 0; i < LS_BLOCKS; ++i) {   // fixed order
      const float2 p = part[i];
      const float mm = fmaxf(m, p.x);
      s = s * expf(m - mm) + p.y * expf(p.x - mm);
      m = mm;
    }
    lse[0] = m + logf(s);
  }
}

__global__ void __launch_bounds__(256)
ls_write(const float* __restrict__ logits, const float* __restrict__ lse,
         float* __restrict__ out, int n)
{
  const float l = lse[0];
  for (int i = blockIdx.x * 256 + threadIdx.x; i < n; i += 256 * LS_BLOCKS)
    out[i] = logits[i] - l;
}

// ---------------------------------------------------------------------------
// Launch
// ---------------------------------------------------------------------------
extern "C" void kernel_launch(void* const* d_in, const int* in_sizes, int n_in,
                              void* d_out, int out_size, void* d_ws, size_t ws_size,
                              hipStream_t stream)
{
  const int*   input   = (const int*)  d_in[0];
  const float* hidden  = (const float*)d_in[1];
  const float* enc_out = (const float*)d_in[2];
  // d_in[3] encoder_hiddens: unused by the reference
  const float* emb     = (const float*)d_in[4];
  const float* attn_W  = (const float*)d_in[5];
  const float* attn_b  = (const float*)d_in[6];
  const float* comb_W  = (const float*)d_in[7];
  const float* comb_b  = (const float*)d_in[8];
  const float* w_ih    = (const float*)d_in[9];
  const float* w_hh    = (const float*)d_in[10];
  const float* b_ih    = (const float*)d_in[11];
  const float* b_hh    = (const float*)d_in[12];
  const float* out_W   = (const float*)d_in[13];
  const float* out_b   = (const float*)d_in[14];

  float* out = (float*)d_out;          // [V] log-softmax, then [H] h_new
  float* ws  = (float*)d_ws;

  float*  cat2   = ws;                 // 2048: [embedded | attn_applied]
  float*  xcomb  = ws + 2048;          // 1024
  float*  gi     = ws + 3072;          // 3072
  float*  gh     = ws + 6144;          // 3072
  float*  hnew   = ws + 9216;          // 1024
  float*  logits = ws + 10240;         // 50257  (ends at 60497)
  float2* part   = (float2*)(ws + 60512); // 128 float2 = 256 floats
  float*  lse    = ws + 60800;         // 1

  attn_kernel<<<1, 256, 0, stream>>>(input, hidden, enc_out, emb, attn_W, attn_b, cat2);

  // x = relu(comb_W @ cat2 + comb_b): M=1024, K=2048, 64 rows/block -> 16 blocks
  gemv_wmma<<<16, 128, 0, stream>>>(comb_W, cat2, comb_b, xcomb, H_DIM, 2 * H_DIM, 1);

  // fused GRU gate GEMVs: virtual M=6144 -> 384 tiles -> 96 blocks
  gates_gemv<<<96, 128, 0, stream>>>(w_ih, xcomb, b_ih, w_hh, hidden, b_hh, gi, gh);

  gru_kernel<<<4, 256, 0, stream>>>(gi, gh, hidden, hnew, out + V_DIM);

  // logits = out_W @ hnew + out_b: 3142 tiles -> 786 blocks
  gemv_wmma<<<786, 128, 0, stream>>>(out_W, hnew, out_b, logits, V_DIM, H_DIM, 0);

  ls_partial<<<LS_BLOCKS, 256, 0, stream>>>(logits, V_DIM, part);
  ls_final<<<1, 32, 0, stream>>>(part, lse);
  ls_write<<<LS_BLOCKS, 256, 0, stream>>>(logits, lse, out, V_DIM);
}